// GATEncoder_25555055411318
// MI455X (gfx1250) — compile-verified
//
#include <hip/hip_runtime.h>
#include <math.h>

#define NNODES 50000
#define IN_C   128
#define HID    64
#define HEADS  4
#define H1DIM  256   // HEADS*HID
#define OUT_C  64

typedef float v2f __attribute__((ext_vector_type(2)));
typedef float v8f __attribute__((ext_vector_type(8)));

// ---- order-preserving float <-> uint mapping for atomicMax on floats ----
__device__ __forceinline__ unsigned floatFlip(float f) {
  unsigned u = __float_as_uint(f);
  return (u & 0x80000000u) ? ~u : (u | 0x80000000u);
}
__device__ __forceinline__ float floatUnflip(unsigned u) {
  return __uint_as_float((u & 0x80000000u) ? (u & 0x7FFFFFFFu) : ~u);
}

__global__ void fill_u32_kernel(unsigned* __restrict__ p, unsigned v, int n) {
  int t = blockIdx.x * blockDim.x + threadIdx.x;
  if (t < n) p[t] = v;
}

// ---------------------------------------------------------------------------
// Dense GEMM C[M,NW] = A[M,K] @ B[K,NW] using V_WMMA_F32_16X16X4_F32.
// One wave per 16x16 output tile; wave-uniform early exit keeps EXEC all-1s.
// A layout per ISA: lanes 0-15 hold K=0,1; lanes 16-31 hold K=2,3 (per row).
// B layout mirrored; C/D: VGPR r -> row r (lanes 0-15) / row 8+r (lanes 16-31).
// ---------------------------------------------------------------------------
template <int K, int NW>
__global__ __launch_bounds__(256) void gemm_wmma_f32(
    const float* __restrict__ A, const float* __restrict__ B,
    float* __restrict__ C, int totalTiles) {
  int wave = blockIdx.x * (blockDim.x >> 5) + (threadIdx.x >> 5);
  if (wave >= totalTiles) return;                 // wave-uniform
  int lane = threadIdx.x & 31;
  const int tilesN = NW / 16;
  int tm = wave / tilesN, tn = wave % tilesN;
  int g = lane >> 4;        // 0: K pair {0,1}; 1: K pair {2,3}
  int l = lane & 15;        // row (A) / col (B)

  const float* aptr = A + (size_t)(tm * 16 + l) * K + 2 * g;
  const float* bptr = B + (size_t)(2 * g) * NW + (tn * 16 + l);

  v8f acc = {};
#pragma unroll 4
  for (int k = 0; k < K; k += 4) {
    v2f av; av.x = aptr[0]; av.y = aptr[1];
    v2f bv; bv.x = bptr[0]; bv.y = bptr[NW];
    acc = __builtin_amdgcn_wmma_f32_16x16x4_f32(
        false, av, false, bv, (short)0, acc, false, false);
    aptr += 4;
    bptr += 4 * NW;
  }
  float* cptr = C + (size_t)(tm * 16 + g * 8) * NW + (tn * 16 + l);
#pragma unroll
  for (int r = 0; r < 8; ++r) cptr[(size_t)r * NW] = acc[r];
}

// a_src[n,h] = dot(h[n,h,:], att_src[h,:]);  same for a_dst. ch == 64 always.
__global__ void att_kernel(const float* __restrict__ h,
                           const float* __restrict__ attS,
                           const float* __restrict__ attD,
                           float* __restrict__ aS, float* __restrict__ aD,
                           int heads) {
  int t = blockIdx.x * blockDim.x + threadIdx.x;
  if (t >= NNODES * heads) return;
  int n = t / heads, hd = t - n * heads;
  const float* hp = h + (size_t)n * heads * 64 + hd * 64;
  const float* ws = attS + hd * 64;
  const float* wd = attD + hd * 64;
  float s = 0.f, d = 0.f;
#pragma unroll 8
  for (int c = 0; c < 64; ++c) { float v = hp[c]; s += v * ws[c]; d += v * wd[c]; }
  aS[t] = s; aD[t] = d;
}

// e = leaky_relu(a_src[src]+a_dst[dst], 0.2); segment max via flipped atomicMax
__global__ void edge_score_kernel(const int* __restrict__ ei, int E, int heads,
                                  const float* __restrict__ aS,
                                  const float* __restrict__ aD,
                                  float* __restrict__ esc,
                                  unsigned* __restrict__ mmax) {
  int t = blockIdx.x * blockDim.x + threadIdx.x;
  if (t >= E * heads) return;
  int e = t / heads, hd = t - e * heads;
  int s = ei[e], d = ei[E + e];
  float v = aS[s * heads + hd] + aD[d * heads + hd];
  v = v > 0.f ? v : 0.2f * v;
  esc[t] = v;
  atomicMax(&mmax[d * heads + hd], floatFlip(v));
}

// ex = exp(e - m[dst]); segment sum via atomicAdd
__global__ void edge_exp_kernel(const int* __restrict__ ei, int E, int heads,
                                const unsigned* __restrict__ mmax,
                                float* __restrict__ esc,
                                float* __restrict__ ssum) {
  int t = blockIdx.x * blockDim.x + threadIdx.x;
  if (t >= E * heads) return;
  int e = t / heads, hd = t - e * heads;
  int d = ei[E + e];
  float m = floatUnflip(mmax[d * heads + hd]);
  float ex = expf(esc[t] - m);
  esc[t] = ex;
  atomicAdd(&ssum[d * heads + hd], ex);
}

// acc[dst] += h[src] * alpha; one thread = 4 channels (float4 gather, 4 atomics)
__global__ void aggregate_kernel(const int* __restrict__ ei, int E, int heads,
                                 int ch, const float* __restrict__ esc,
                                 const float* __restrict__ ssum,
                                 const float* __restrict__ hfeat,
                                 float* __restrict__ acc) {
  int perEdge = (heads * ch) >> 2;
  long total = (long)E * perEdge;
  long t = (long)blockIdx.x * blockDim.x + threadIdx.x;
  if (t >= total) return;
  int e  = (int)(t / perEdge);
  int r  = (int)(t - (long)e * perEdge);
  int c0 = r << 2;
  int hd = c0 / ch;
  int s = ei[e], d = ei[E + e];
  float ssv = ssum[d * heads + hd];
  float alpha = esc[e * heads + hd] / (ssv > 0.f ? ssv : 1.f);
  float4 hv = *(const float4*)(hfeat + (size_t)s * heads * ch + c0);
  float* op = acc + (size_t)d * heads * ch + c0;
  atomicAdd(op + 0, hv.x * alpha);
  atomicAdd(op + 1, hv.y * alpha);
  atomicAdd(op + 2, hv.z * alpha);
  atomicAdd(op + 3, hv.w * alpha);
}

__global__ void bias_elu_kernel(const float* __restrict__ acc,
                                const float* __restrict__ b,
                                float* __restrict__ out, int total, int wmask) {
  int t = blockIdx.x * blockDim.x + threadIdx.x;
  if (t >= total) return;
  float v = acc[t] + b[t & wmask];
  out[t] = v > 0.f ? v : expm1f(v);
}

__global__ void bias_kernel(const float* __restrict__ acc,
                            const float* __restrict__ b,
                            float* __restrict__ out, int total, int wmask) {
  int t = blockIdx.x * blockDim.x + threadIdx.x;
  if (t >= total) return;
  out[t] = acc[t] + b[t & wmask];
}

extern "C" void kernel_launch(void* const* d_in, const int* in_sizes, int n_in,
                              void* d_out, int out_size, void* d_ws, size_t ws_size,
                              hipStream_t stream) {
  const float* x   = (const float*)d_in[0];
  const int*   ei  = (const int*)d_in[1];
  const float* W1  = (const float*)d_in[2];
  const float* as1 = (const float*)d_in[3];
  const float* ad1 = (const float*)d_in[4];
  const float* b1  = (const float*)d_in[5];
  const float* W2  = (const float*)d_in[6];
  const float* as2 = (const float*)d_in[7];
  const float* ad2 = (const float*)d_in[8];
  const float* b2  = (const float*)d_in[9];
  float* out = (float*)d_out;
  const int E = in_sizes[1] / 2;

  // ---- workspace layout (floats) ----
  float* w = (float*)d_ws;
  size_t off = 0;
  float*    h1   = w + off; off += (size_t)NNODES * H1DIM;   // x @ W1, later elu(h)
  float*    hsum = w + off; off += (size_t)NNODES * H1DIM;   // layer1 aggregation
  float*    h2   = w + off; off += (size_t)NNODES * OUT_C;   // h @ W2
  float*    o2   = w + off; off += (size_t)NNODES * OUT_C;   // layer2 aggregation
  float*    aS1  = w + off; off += (size_t)NNODES * HEADS;
  float*    aD1  = w + off; off += (size_t)NNODES * HEADS;
  unsigned* m1   = (unsigned*)(w + off); off += (size_t)NNODES * HEADS;
  float*    s1   = w + off; off += (size_t)NNODES * HEADS;
  float*    aS2  = w + off; off += (size_t)NNODES;
  float*    aD2  = w + off; off += (size_t)NNODES;
  unsigned* m2   = (unsigned*)(w + off); off += (size_t)NNODES;
  float*    s2   = w + off; off += (size_t)NNODES;
  float*    ex1  = w + off; off += (size_t)E * HEADS;
  float*    ex2  = w + off; off += (size_t)E;

  const unsigned FLIP_NEG_INF = 0x007FFFFFu;  // floatFlip(-inf)

  // ---- per-call init (accumulators must be reset every launch) ----
  hipMemsetAsync(hsum, 0, (size_t)NNODES * H1DIM * sizeof(float), stream);
  hipMemsetAsync(o2,   0, (size_t)NNODES * OUT_C * sizeof(float), stream);
  hipMemsetAsync(s1,   0, (size_t)NNODES * HEADS * sizeof(float), stream);
  hipMemsetAsync(s2,   0, (size_t)NNODES * sizeof(float), stream);
  fill_u32_kernel<<<(NNODES * HEADS + 255) / 256, 256, 0, stream>>>(m1, FLIP_NEG_INF, NNODES * HEADS);
  fill_u32_kernel<<<(NNODES + 255) / 256, 256, 0, stream>>>(m2, FLIP_NEG_INF, NNODES);

  // ---- layer 1 ----
  const int tiles1 = (NNODES / 16) * (H1DIM / 16);   // 3125 * 16 = 50000 waves
  gemm_wmma_f32<IN_C, H1DIM><<<(tiles1 + 7) / 8, 256, 0, stream>>>(x, W1, h1, tiles1);
  att_kernel<<<(NNODES * HEADS + 255) / 256, 256, 0, stream>>>(h1, as1, ad1, aS1, aD1, HEADS);
  edge_score_kernel<<<(E * HEADS + 255) / 256, 256, 0, stream>>>(ei, E, HEADS, aS1, aD1, ex1, m1);
  edge_exp_kernel<<<(E * HEADS + 255) / 256, 256, 0, stream>>>(ei, E, HEADS, m1, ex1, s1);
  long agg1 = (long)E * ((HEADS * HID) >> 2);
  aggregate_kernel<<<(unsigned)((agg1 + 255) / 256), 256, 0, stream>>>(ei, E, HEADS, HID, ex1, s1, h1, hsum);
  bias_elu_kernel<<<(NNODES * H1DIM + 255) / 256, 256, 0, stream>>>(hsum, b1, h1, NNODES * H1DIM, H1DIM - 1);

  // ---- layer 2 (heads=1, concat=False -> mean over 1 head == identity) ----
  const int tiles2 = (NNODES / 16) * (OUT_C / 16);   // 3125 * 4 = 12500 waves
  gemm_wmma_f32<H1DIM, OUT_C><<<(tiles2 + 7) / 8, 256, 0, stream>>>(h1, W2, h2, tiles2);
  att_kernel<<<(NNODES + 255) / 256, 256, 0, stream>>>(h2, as2, ad2, aS2, aD2, 1);
  edge_score_kernel<<<(E + 255) / 256, 256, 0, stream>>>(ei, E, 1, aS2, aD2, ex2, m2);
  edge_exp_kernel<<<(E + 255) / 256, 256, 0, stream>>>(ei, E, 1, m2, ex2, s2);
  long agg2 = (long)E * (OUT_C >> 2);
  aggregate_kernel<<<(unsigned)((agg2 + 255) / 256), 256, 0, stream>>>(ei, E, 1, OUT_C, ex2, s2, h2, o2);
  bias_kernel<<<(NNODES * OUT_C + 255) / 256, 256, 0, stream>>>(o2, b2, out, NNODES * OUT_C, OUT_C - 1);
}